// DepthwiseCausalConv1d_266287972534
// MI455X (gfx1250) — compile-verified
//
#include <hip/hip_runtime.h>

// Causal depthwise conv1d, K=4, dilation=1, fp32, [B,T,C] layout.
// y[b,t,c] = w0*x[t-3] + w1*x[t-2] + w2*x[t-1] + w3*x[t]   (per channel c)
//
// Roofline: 268 MFLOP vs 268 MB minimum traffic -> ~1 FLOP/B -> pure HBM
// streaming (~11.5 us at 23.3 TB/s). No GEMM structure (per-channel taps),
// so the CDNA5 path to use is the async global->LDS copy engine
// (GLOBAL_LOAD_ASYNC_TO_LDS_B128 + ASYNCcnt), double-buffered, with direct
// NT stores for y. Each thread consumes only the LDS bytes it staged itself
// (depthwise => channel-private, history is along t within a lane), so no
// workgroup barriers are required: s_wait_asynccnt fully orders copy->read.

typedef __attribute__((ext_vector_type(4))) float v4f;
typedef __attribute__((ext_vector_type(4))) int   v4i;

#define CONV_K  4
#define T_TILE  128          // time rows per thread (halo overhead 3/128)
#define CHUNK   8            // rows staged per async batch
#define NCHUNK  (T_TILE / CHUNK)
#define BLOCK   256          // = C/4 channel quads; 8 wave32 waves

#if defined(__has_builtin)
#  if __has_builtin(__builtin_amdgcn_global_load_async_to_lds_b128)
#    define HAVE_ASYNC_LDS 1
#  else
#    define HAVE_ASYNC_LDS 0
#  endif
#  if __has_builtin(__builtin_amdgcn_s_wait_asynccnt)
#    define HAVE_WAIT_ASYNC 1
#  else
#    define HAVE_WAIT_ASYNC 0
#  endif
#else
#  define HAVE_ASYNC_LDS 0
#  define HAVE_WAIT_ASYNC 0
#endif

// b128 async copy moves one 4x32-bit vector per lane: pointee type is v4i.
typedef __attribute__((address_space(1))) v4i glb_v4i_t;   // global (AS1)
typedef __attribute__((address_space(3))) v4i lds_v4i_t;   // LDS (AS3)

__device__ __forceinline__ void wait_asynccnt_8() {
#if HAVE_WAIT_ASYNC
    __builtin_amdgcn_s_wait_asynccnt(8);
    asm volatile("" ::: "memory");        // keep LDS reads below the wait
#else
    asm volatile("s_wait_asynccnt 0x8" ::: "memory");
#endif
}
__device__ __forceinline__ void wait_asynccnt_0() {
#if HAVE_WAIT_ASYNC
    __builtin_amdgcn_s_wait_asynccnt(0);
    asm volatile("" ::: "memory");
#else
    asm volatile("s_wait_asynccnt 0x0" ::: "memory");
#endif
}

#if HAVE_ASYNC_LDS
__global__ __launch_bounds__(BLOCK, 1) void dwconv_causal_k4_async(
    const float* __restrict__ x,
    const float* __restrict__ W,
    float* __restrict__ y,
    int T, int C)
{
    __shared__ v4f lds[2 * CHUNK * BLOCK];        // 64 KB of the 320 KB WGP LDS

    const int cg   = threadIdx.x;                 // channel quad id
    const int tile = blockIdx.x;                  // T tile
    const int b    = blockIdx.y;                  // batch

    const int c  = cg << 2;
    const int t0 = tile * T_TILE;
    const int sV = C >> 2;                        // row stride in v4f units

    const v4f* __restrict__ xv = (const v4f*)(x + ((size_t)b * T + (size_t)t0) * C + c);
    v4f*       __restrict__ yv = (v4f*)      (y + ((size_t)b * T + (size_t)t0) * C + c);

    const v4f w0 = *(const v4f*)(W + 0 * C + c);
    const v4f w1 = *(const v4f*)(W + 1 * C + c);
    const v4f w2 = *(const v4f*)(W + 2 * C + c);
    const v4f w3 = *(const v4f*)(W + 3 * C + c);

    v4f h0 = {0.0f, 0.0f, 0.0f, 0.0f};
    v4f h1 = h0, h2 = h0;
    if (t0 > 0) {                                 // t0 == 0 or >= T_TILE (>3)
        h0 = xv[-3 * sV];
        h1 = xv[-2 * sV];
        h2 = xv[-1 * sV];
    }

    // Issue CHUNK per-lane async b128 copies for chunk `ch` into its LDS buffer.
    auto issue = [&](int ch) {
        const v4f* g = xv + (size_t)(ch * CHUNK) * sV;
        v4f*       l = &lds[(ch & 1) * (CHUNK * BLOCK) + cg];
        #pragma unroll
        for (int r = 0; r < CHUNK; ++r) {
            __builtin_amdgcn_global_load_async_to_lds_b128(
                (glb_v4i_t*)(g + (size_t)r * sV),
                (lds_v4i_t*)(l + r * BLOCK),
                /*imm offset*/0, /*cpol*/0);
        }
    };

    issue(0);                                     // prologue
    #pragma unroll 1
    for (int ch = 0; ch < NCHUNK; ++ch) {
        if (ch + 1 < NCHUNK) {
            issue(ch + 1);                        // keep next chunk in flight
            wait_asynccnt_8();                    // in-order: chunk `ch` landed
        } else {
            wait_asynccnt_0();                    // drain tail
        }

        const v4f* l = &lds[(ch & 1) * (CHUNK * BLOCK) + cg];
        #pragma unroll
        for (int r = 0; r < CHUNK; ++r) {
            v4f v = l[r * BLOCK];                 // ds_load_b128
            v4f o = w0 * h0 + w1 * h1 + w2 * h2 + w3 * v;
            __builtin_nontemporal_store(o, yv + (size_t)(ch * CHUNK + r) * sV);
            h0 = h1; h1 = h2; h2 = v;
        }
    }
}
#else  // ------- fallback: register rolling-window streaming kernel -------
__global__ __launch_bounds__(BLOCK, 1) void dwconv_causal_k4_async(
    const float* __restrict__ x,
    const float* __restrict__ W,
    float* __restrict__ y,
    int T, int C)
{
    const int cg   = threadIdx.x;
    const int tile = blockIdx.x;
    const int b    = blockIdx.y;

    const int c  = cg << 2;
    const int t0 = tile * T_TILE;
    const int sV = C >> 2;

    const v4f* __restrict__ xv = (const v4f*)(x + ((size_t)b * T + (size_t)t0) * C + c);
    v4f*       __restrict__ yv = (v4f*)      (y + ((size_t)b * T + (size_t)t0) * C + c);

    const v4f w0 = *(const v4f*)(W + 0 * C + c);
    const v4f w1 = *(const v4f*)(W + 1 * C + c);
    const v4f w2 = *(const v4f*)(W + 2 * C + c);
    const v4f w3 = *(const v4f*)(W + 3 * C + c);

    v4f h0 = {0.0f, 0.0f, 0.0f, 0.0f};
    v4f h1 = h0, h2 = h0;
    if (t0 > 0) {
        h0 = xv[-3 * sV];
        h1 = xv[-2 * sV];
        h2 = xv[-1 * sV];
    }

    #pragma unroll 1
    for (int i = 0; i < T_TILE; i += CHUNK) {
        v4f v[CHUNK];
        #pragma unroll
        for (int j = 0; j < CHUNK; ++j)
            v[j] = __builtin_nontemporal_load(&xv[(size_t)(i + j) * sV]);
        #pragma unroll
        for (int j = 0; j < CHUNK; ++j) {
            v4f o = w0 * h0 + w1 * h1 + w2 * h2 + w3 * v[j];
            __builtin_nontemporal_store(o, &yv[(size_t)(i + j) * sV]);
            h0 = h1; h1 = h2; h2 = v[j];
        }
    }
}
#endif

extern "C" void kernel_launch(void* const* d_in, const int* in_sizes, int n_in,
                              void* d_out, int out_size, void* d_ws, size_t ws_size,
                              hipStream_t stream)
{
    const float* x = (const float*)d_in[0];   // [B, T, C] fp32
    const float* W = (const float*)d_in[1];   // [K, 1, C] fp32
    float*       y = (float*)d_out;           // [B, T, C] fp32

    const int C = in_sizes[1] / CONV_K;       // 1024
    const int T = 4096;
    const int B = (in_sizes[0] / C) / T;      // 8

    dim3 grid((unsigned)(T / T_TILE), (unsigned)B);   // 32 x 8 blocks
    dim3 block((unsigned)(C / 4));                    // 256 threads

    hipLaunchKernelGGL(dwconv_causal_k4_async, grid, block, 0, stream,
                       x, W, y, T, C);
}